// Rank_Loss_7438883356888
// MI455X (gfx1250) — compile-verified
//
#include <hip/hip_runtime.h>
#include <math.h>

// ---- problem constants (from reference) ----
#define N_TOT        4096
#define D_FEAT_      2048
#define P_ROWS       1024
#define NUM_INST_    4
#define NUM_CLASSES_ 1024
#define MARGIN2_     0.3f
#define THR_ADD      (0.3f / 3.0f)   // MARGIN2/DIVIDE

typedef __attribute__((ext_vector_type(2))) float v2f;
typedef __attribute__((ext_vector_type(8))) float v8f;
typedef __attribute__((ext_vector_type(4))) int   v4i;

// ---- workspace layout (float offsets) ----
#define WS_SQ       0            // 4096 floats
#define WS_ROWLOSS  4096         // 1024 floats
#define WS_XENT     8192         // 4096 floats
#define WS_S42      16384        // 2048 floats
#define WS_S43      (16384+2048) // 2048 floats
#define WS_DIST     (1<<18)      // 1024*4096 floats at byte offset 1MB

// ---- CDNA5 async global->LDS path (guarded; falls back to sync copies) ----
#if defined(__AMDGCN__) && __has_builtin(__builtin_amdgcn_global_load_async_to_lds_b128)
#define HAS_ASYNC_LDS 1
#else
#define HAS_ASYNC_LDS 0
#endif

#if HAS_ASYNC_LDS
// param 1: v4i addrspace(1)* (global source), param 2: v4i addrspace(3)* (LDS dest),
// then imm offset, imm cpol (per clang diagnostic from probe round).
#define ASYNC_CP16(gsrc, ldst)                                              \
    __builtin_amdgcn_global_load_async_to_lds_b128(                         \
        (__attribute__((address_space(1))) v4i*)(gsrc),                     \
        (__attribute__((address_space(3))) v4i*)(ldst), 0, 0)
__device__ __forceinline__ void wait_async_lds() {
#if __has_builtin(__builtin_amdgcn_s_wait_asynccnt)
    __builtin_amdgcn_s_wait_asynccnt(0);
#else
    asm volatile("s_wait_asynccnt 0" ::: "memory");
#endif
}
#endif

// ------------------------------------------------------------------
// K1: sq[i] = sum_d x[i,d]^2
// ------------------------------------------------------------------
__global__ __launch_bounds__(256) void k_rowsq(const float* __restrict__ x,
                                               float* __restrict__ sq) {
    __shared__ float red[256];
    int row = blockIdx.x;
    int t = threadIdx.x;
    const float4* xr = (const float4*)(x + (size_t)row * D_FEAT_);
    float4 a = xr[t];
    float4 b = xr[t + 256];
    float s = a.x*a.x + a.y*a.y + a.z*a.z + a.w*a.w
            + b.x*b.x + b.y*b.y + b.z*b.z + b.w*b.w;
    red[t] = s; __syncthreads();
    for (int off = 128; off > 0; off >>= 1) {
        if (t < off) red[t] += red[t + off];
        __syncthreads();
    }
    if (t == 0) sq[row] = red[0];
}

// ------------------------------------------------------------------
// K2: fused GEMM (f32 WMMA 16x16x4) + distance epilogue, with
// double-buffered async global->LDS staging.
//   dist[i,j] = sqrt(max(sq[i]+sq[j]-2*<x_i,x_j>, 1e-12)), i<1024, j<4096
// Block: 256 thr (8 waves), tile 64x64, K-step 32.
// Fragment layouts per CDNA5 ISA 7.12.2:
//   A 16x4 : lane L -> M=L%16, VGPR v -> K = v + 2*(L/16)
//   B 4x16 : lane L -> N=L%16, VGPR v -> K = v + 2*(L/16)
//   C/D    : lane L, VGPR v -> M = v + 8*(L/16), N = L%16
// ------------------------------------------------------------------
#define TM 64
#define TN 64
#define TK 32
#define LDP 36   // padded LDS row stride: 144B (16B aligned), conflict-free frags

__global__ __launch_bounds__(256) void k_gemm_dist(const float* __restrict__ x,
                                                   const float* __restrict__ sq,
                                                   float* __restrict__ dist) {
    __shared__ float As[2][TM * LDP];
    __shared__ float Bs[2][TN * LDP];

    const int lane  = threadIdx.x & 31;
    const int wave  = threadIdx.x >> 5;
    const int wr    = wave >> 1;        // 0..3
    const int wc    = wave & 1;         // 0..1
    const int lhalf = lane >> 4;        // 0 or 1
    const int lmod  = lane & 15;

    const int rowBase = blockIdx.y * TM;   // 0..960   (A rows: first 1024 only)
    const int colBase = blockIdx.x * TN;   // 0..4032  (B rows of x)

    // precomputed per-thread staging addresses: rows r0 and r0+32, k-offset kq
    const int r0 = threadIdx.x >> 3;        // 0..31
    const int kq = (threadIdx.x & 7) << 2;  // 0,4,...,28
    const float* gA0 = x + (size_t)(rowBase + r0)      * D_FEAT_ + kq;
    const float* gA1 = x + (size_t)(rowBase + r0 + 32) * D_FEAT_ + kq;
    const float* gB0 = x + (size_t)(colBase + r0)      * D_FEAT_ + kq;
    const float* gB1 = x + (size_t)(colBase + r0 + 32) * D_FEAT_ + kq;
    const int l0 = r0 * LDP + kq;
    const int l1 = (r0 + 32) * LDP + kq;

    auto stage = [&](int kb, int b) {
#if HAS_ASYNC_LDS
        ASYNC_CP16(gA0 + kb, &As[b][l0]);
        ASYNC_CP16(gA1 + kb, &As[b][l1]);
        ASYNC_CP16(gB0 + kb, &Bs[b][l0]);
        ASYNC_CP16(gB1 + kb, &Bs[b][l1]);
#else
        *(float4*)(&As[b][l0]) = *(const float4*)(gA0 + kb);
        *(float4*)(&As[b][l1]) = *(const float4*)(gA1 + kb);
        *(float4*)(&Bs[b][l0]) = *(const float4*)(gB0 + kb);
        *(float4*)(&Bs[b][l1]) = *(const float4*)(gB1 + kb);
#endif
    };

    v8f acc0 = {};
    v8f acc1 = {};

    // prologue: stage first tile
    stage(0, 0);
#if HAS_ASYNC_LDS
    wait_async_lds();
#endif
    __syncthreads();

    int buf = 0;
    const int arow  = (16 * wr + lmod) * LDP;
    const int brow0 = (32 * wc + lmod) * LDP;
    const int brow1 = (32 * wc + 16 + lmod) * LDP;

    for (int kb = 0; kb < D_FEAT_; kb += TK) {
        const int nkb = kb + TK;
        if (nkb < D_FEAT_) stage(nkb, buf ^ 1);   // overlap next-tile DMA

        #pragma unroll
        for (int kk = 0; kk < TK; kk += 4) {
            const int ak = kk + 2 * lhalf;
            v2f a, b0, b1;
            const float* ap  = &As[buf][arow + ak];
            a[0]  = ap[0];  a[1]  = ap[1];
            const float* bp0 = &Bs[buf][brow0 + ak];
            b0[0] = bp0[0]; b0[1] = bp0[1];
            const float* bp1 = &Bs[buf][brow1 + ak];
            b1[0] = bp1[0]; b1[1] = bp1[1];

            acc0 = __builtin_amdgcn_wmma_f32_16x16x4_f32(
                false, a, false, b0, (short)0, acc0, false, false);
            acc1 = __builtin_amdgcn_wmma_f32_16x16x4_f32(
                false, a, false, b1, (short)0, acc1, false, false);
        }

#if HAS_ASYNC_LDS
        wait_async_lds();      // my next-buffer DMAs have landed
#endif
        __syncthreads();       // everyone done reading buf + writing buf^1
        buf ^= 1;
    }

    // epilogue: distances
    const int gj0 = colBase + 32 * wc + lmod;
    const float sqj0 = sq[gj0];
    const float sqj1 = sq[gj0 + 16];
    #pragma unroll
    for (int v = 0; v < 8; ++v) {
        int gi = rowBase + 16 * wr + v + 8 * lhalf;
        float sqi = sq[gi];
        float d2a = sqi + sqj0 - 2.0f * acc0[v];
        float d2b = sqi + sqj1 - 2.0f * acc1[v];
        dist[(size_t)gi * N_TOT + gj0]      = sqrtf(fmaxf(d2a, 1e-12f));
        dist[(size_t)gi * N_TOT + gj0 + 16] = sqrtf(fmaxf(d2b, 1e-12f));
    }
}

// ------------------------------------------------------------------
// K3: per-row triplet term. One block per row i (i < 1024).
// Single global pass: row cached in LDS (16KB), softmax pass reads LDS.
// ------------------------------------------------------------------
__global__ __launch_bounds__(256) void k_triplet(const float* __restrict__ dist,
                                                 float* __restrict__ rowloss) {
    __shared__ float drow_s[N_TOT];   // 16KB row cache
    __shared__ float red[256];
    __shared__ float red2[256];
    int i = blockIdx.x;
    int t = threadIdx.x;
    const float* drow = dist + (size_t)i * N_TOT;
    int posBase = i & ~(NUM_INST_ - 1);

    // pass 1: cache row + hardest-negative min over j with j//4 != i//4
    float m = 3.0e38f;
    for (int j = t; j < N_TOT; j += 256) {
        float d = drow[j];
        drow_s[j] = d;
        if ((j & ~3) != posBase) m = fminf(m, d);
    }
    red[t] = m; __syncthreads();
    for (int off = 128; off > 0; off >>= 1) {
        if (t < off) red[t] = fminf(red[t], red[t + off]);
        __syncthreads();
    }
    float negmin = red[0];
    __syncthreads();

    // pass 2 (LDS): softmax(-d) weighted mean over selected negatives
    float thr = negmin + THR_ADD;
    float sw = 0.0f, swd = 0.0f;
    for (int j = t; j < N_TOT; j += 256) {
        float d = drow_s[j];
        if ((j & ~3) != posBase && d < thr) {
            float e = expf(negmin - d);   // stable: d >= negmin
            sw += e; swd += e * d;
        }
    }
    red[t] = sw; red2[t] = swd; __syncthreads();
    for (int off = 128; off > 0; off >>= 1) {
        if (t < off) { red[t] += red[t + off]; red2[t] += red2[t + off]; }
        __syncthreads();
    }

    if (t == 0) {
        float negative2 = red2[0] / red[0];
        int self = i & (NUM_INST_ - 1);
        float p[3]; int c = 0;
        for (int k = 0; k < NUM_INST_; ++k)
            if (k != self) p[c++] = drow_s[posBase + k];
        float mx = fmaxf(p[0], fmaxf(p[1], p[2]));
        float e0 = expf(p[0] - mx), e1 = expf(p[1] - mx), e2 = expf(p[2] - mx);
        float positive2 = (e0 * p[0] + e1 * p[1] + e2 * p[2]) / (e0 + e1 + e2);
        rowloss[i] = fmaxf(MARGIN2_ - (negative2 - positive2), 0.0f);
    }
}

// ------------------------------------------------------------------
// K4: cross-entropy per row: lse(row) - row[target]
// ------------------------------------------------------------------
__global__ __launch_bounds__(256) void k_xent(const float* __restrict__ cls,
                                              const int* __restrict__ tgt,
                                              float* __restrict__ xrow) {
    __shared__ float red[256];
    int r = blockIdx.x, t = threadIdx.x;
    const float* row = cls + (size_t)r * NUM_CLASSES_;
    float m = -3.0e38f;
    for (int c = t; c < NUM_CLASSES_; c += 256) m = fmaxf(m, row[c]);
    red[t] = m; __syncthreads();
    for (int off = 128; off > 0; off >>= 1) {
        if (t < off) red[t] = fmaxf(red[t], red[t + off]);
        __syncthreads();
    }
    float mx = red[0]; __syncthreads();
    float s = 0.0f;
    for (int c = t; c < NUM_CLASSES_; c += 256) s += expf(row[c] - mx);
    red[t] = s; __syncthreads();
    for (int off = 128; off > 0; off >>= 1) {
        if (t < off) red[t] += red[t + off];
        __syncthreads();
    }
    if (t == 0) xrow[r] = (logf(red[0]) + mx) - row[tgt[r]];
}

// ------------------------------------------------------------------
// K5: side-loss partial sums: ||l4-l2||^2 and ||l4-l3||^2, per-block
// ------------------------------------------------------------------
__global__ __launch_bounds__(256) void k_side(const float* __restrict__ l2,
                                              const float* __restrict__ l3,
                                              const float* __restrict__ l4,
                                              float* __restrict__ p42,
                                              float* __restrict__ p43) {
    __shared__ float r42[256];
    __shared__ float r43[256];
    int t = threadIdx.x;
    size_t base = (size_t)blockIdx.x * 2048;
    float s42 = 0.0f, s43 = 0.0f;
    #pragma unroll
    for (int k = 0; k < 2; ++k) {
        size_t idx = base + (size_t)k * 1024 + (size_t)t * 4;
        float4 a4 = *(const float4*)(l4 + idx);
        float4 a2 = *(const float4*)(l2 + idx);
        float4 a3 = *(const float4*)(l3 + idx);
        float d;
        d = a4.x - a2.x; s42 += d * d;  d = a4.y - a2.y; s42 += d * d;
        d = a4.z - a2.z; s42 += d * d;  d = a4.w - a2.w; s42 += d * d;
        d = a4.x - a3.x; s43 += d * d;  d = a4.y - a3.y; s43 += d * d;
        d = a4.z - a3.z; s43 += d * d;  d = a4.w - a3.w; s43 += d * d;
    }
    r42[t] = s42; r43[t] = s43; __syncthreads();
    for (int off = 128; off > 0; off >>= 1) {
        if (t < off) { r42[t] += r42[t + off]; r43[t] += r43[t + off]; }
        __syncthreads();
    }
    if (t == 0) { p42[blockIdx.x] = r42[0]; p43[blockIdx.x] = r43[0]; }
}

// ------------------------------------------------------------------
// K6: final deterministic combine (single block)
// ------------------------------------------------------------------
__global__ __launch_bounds__(256) void k_final(const float* __restrict__ ws,
                                               float* __restrict__ out) {
    __shared__ float red[256];
    int t = threadIdx.x;
    const float* rowloss = ws + WS_ROWLOSS;
    const float* xrow    = ws + WS_XENT;
    const float* p42     = ws + WS_S42;
    const float* p43     = ws + WS_S43;

    float s = 0.0f;
    for (int k = 0; k < 4; ++k) s += rowloss[t + 256 * k];
    red[t] = s; __syncthreads();
    for (int off = 128; off > 0; off >>= 1) { if (t < off) red[t] += red[t+off]; __syncthreads(); }
    float trip = red[0] / (float)P_ROWS; __syncthreads();

    s = 0.0f;
    for (int k = 0; k < 16; ++k) s += xrow[t + 256 * k];
    red[t] = s; __syncthreads();
    for (int off = 128; off > 0; off >>= 1) { if (t < off) red[t] += red[t+off]; __syncthreads(); }
    float xent = red[0] / (float)N_TOT; __syncthreads();

    s = 0.0f;
    for (int k = 0; k < 8; ++k) s += p42[t + 256 * k];
    red[t] = s; __syncthreads();
    for (int off = 128; off > 0; off >>= 1) { if (t < off) red[t] += red[t+off]; __syncthreads(); }
    float s42 = red[0]; __syncthreads();

    s = 0.0f;
    for (int k = 0; k < 8; ++k) s += p43[t + 256 * k];
    red[t] = s; __syncthreads();
    for (int off = 128; off > 0; off >>= 1) { if (t < off) red[t] += red[t+off]; __syncthreads(); }
    float s43 = red[0];

    if (t == 0) {
        // loss = ALPHA*triplet2 + GAMMA*xent + THETA*(loss42 + loss43)
        out[0] = trip + 0.5f * xent + 0.1f * (sqrtf(s42) + sqrtf(s43));
    }
}

// ------------------------------------------------------------------
extern "C" void kernel_launch(void* const* d_in, const int* in_sizes, int n_in,
                              void* d_out, int out_size, void* d_ws, size_t ws_size,
                              hipStream_t stream) {
    const float* cls = (const float*)d_in[0];
    const float* l2  = (const float*)d_in[1];
    const float* l3  = (const float*)d_in[2];
    const float* l4  = (const float*)d_in[3];
    const float* x   = (const float*)d_in[4];
    const int*   tgt = (const int*)d_in[5];

    float* ws      = (float*)d_ws;
    float* sq      = ws + WS_SQ;
    float* rowloss = ws + WS_ROWLOSS;
    float* xrow    = ws + WS_XENT;
    float* p42     = ws + WS_S42;
    float* p43     = ws + WS_S43;
    float* dist    = ws + WS_DIST;

    k_rowsq<<<N_TOT, 256, 0, stream>>>(x, sq);

    dim3 g2(N_TOT / TN, P_ROWS / TM);   // (64, 16)
    k_gemm_dist<<<g2, 256, 0, stream>>>(x, sq, dist);

    k_triplet<<<P_ROWS, 256, 0, stream>>>(dist, rowloss);
    k_xent<<<N_TOT, 256, 0, stream>>>(cls, tgt, xrow);
    k_side<<<2048, 256, 0, stream>>>(l2, l3, l4, p42, p43);
    k_final<<<1, 256, 0, stream>>>(ws, (float*)d_out);
}